// ImageMoE_89361089561040
// MI455X (gfx1250) — compile-verified
//
#include <hip/hip_runtime.h>
#include <math.h>

typedef float v2f __attribute__((ext_vector_type(2)));
typedef float v8f __attribute__((ext_vector_type(8)));

#define NTOK 6272   // B*T = 32*196
#define DMODEL 512

__device__ __forceinline__ v8f wmma4(v2f a, v2f b, v8f c) {
  // V_WMMA_F32_16X16X4_F32 : D = A(16x4 f32) * B(4x16 f32) + C(16x16 f32)
  return __builtin_amdgcn_wmma_f32_16x16x4_f32(false, a, false, b, (short)0, c, false, false);
}

__device__ __forceinline__ float posenc_val(int t, int c) {
  const float kln = 0.01798894602f; // ln(10000)/512
  int i2 = c & ~1;
  float ang = (float)t * expf(-(float)i2 * kln);
  return (c & 1) ? cosf(ang) : sinf(ang);
}

// ---------------------------------------------------------------------------
// Generic fp32 WMMA GEMM: C[M,N] = A[M,K] @ B[K,N] (+bias) (+posenc)
// amode: 0 = plain row-major A, 1 = patchify from image x [32,1,224,224]
// epilogue: 0 = none, 1 = +bias, 2 = +bias +posenc(row%196, col)
// Block 256 (8 waves), tile 64x64. Wave: mw=wid&3 row subtile, nwav=wid>>2.
// ---------------------------------------------------------------------------
__global__ __launch_bounds__(256) void gemm_wmma_kernel(
    const float* __restrict__ A, const float* __restrict__ Bw,
    const float* __restrict__ bias, float* __restrict__ C,
    int M, int N, int Kd, int epilogue, int amode)
{
  __shared__ float As[64][20];   // stride 20: banks (20m mod 64) distinct per half
  __shared__ float Bs[16][72];   // stride 72: half-wave rows land in disjoint banks
  int tid = threadIdx.x;
  int m0 = blockIdx.y * 64, n0 = blockIdx.x * 64;
  int wid = tid >> 5, lane = tid & 31, half = lane >> 4, l15 = lane & 15;
  int mw = wid & 3, nwav = wid >> 2;
  v8f acc0 = {0,0,0,0,0,0,0,0};
  v8f acc1 = {0,0,0,0,0,0,0,0};
  int ra = tid >> 2, ja = (tid & 3) << 2;     // A: 64 rows x 4 float4
  int rb = tid >> 4, jb = (tid & 15) << 2;    // B: 16 rows x 16 float4

  for (int kt = 0; kt < Kd; kt += 16) {
    __syncthreads();
    float4 av = {0.f, 0.f, 0.f, 0.f};
    int grow = m0 + ra;
    if (grow < M) {
      if (amode == 0) {
        av = *(const float4*)(A + (size_t)grow * Kd + kt + ja);
      } else {
        // patchify: row = b*196 + (gy*14+gx); k = py*16+px ; 16-chunk is contiguous
        int bimg = grow / 196, t = grow % 196;
        int gy = t / 14, gx = t % 14;
        int py = (kt + ja) >> 4, px = (kt + ja) & 15;
        av = *(const float4*)(A + ((size_t)(bimg * 224 + gy * 16 + py)) * 224 + gx * 16 + px);
      }
    }
    *(float4*)(&As[ra][ja]) = av;
    float4 bv = *(const float4*)(Bw + (size_t)(kt + rb) * N + n0 + jb);
    *(float4*)(&Bs[rb][jb]) = bv;
    __syncthreads();
#pragma unroll
    for (int k4 = 0; k4 < 16; k4 += 4) {
      int kk = k4 + 2 * half;
      v2f a;  a.x  = As[mw*16 + l15][kk];          a.y  = As[mw*16 + l15][kk + 1];
      v2f b0; b0.x = Bs[kk][nwav*32 + l15];        b0.y = Bs[kk + 1][nwav*32 + l15];
      v2f b1; b1.x = Bs[kk][nwav*32 + 16 + l15];   b1.y = Bs[kk + 1][nwav*32 + 16 + l15];
      acc0 = wmma4(a, b0, acc0);
      acc1 = wmma4(a, b1, acc1);
    }
  }
#pragma unroll
  for (int v = 0; v < 8; v++) {
    int row = m0 + mw*16 + v + 8*half;
    if (row >= M) continue;
    int c0 = n0 + nwav*32 + l15;
    int c1 = c0 + 16;
    float o0 = acc0[v], o1 = acc1[v];
    if (epilogue >= 1) { o0 += bias[c0]; o1 += bias[c1]; }
    if (epilogue == 2) {
      int t = row % 196;
      o0 += posenc_val(t, c0);
      o1 += posenc_val(t, c1);
    }
    C[(size_t)row * N + c0] = o0;
    C[(size_t)row * N + c1] = o1;
  }
}

// ---------------------------------------------------------------------------
// Causal multi-head attention core: one WG per (b,h). 8 waves, each owns
// q-tiles qt = wid, wid+8. Exact softmax, fp32 WMMA for QK^T and PV.
// ---------------------------------------------------------------------------
__global__ __launch_bounds__(256) void attn_kernel(
    const float* __restrict__ q, const float* __restrict__ k,
    const float* __restrict__ v, float* __restrict__ o)
{
  extern __shared__ float smem[];
  float* Kh = smem;                 // [208][68]
  float* Vs = Kh + 208 * 68;        // [208][72]
  float* Qw = Vs + 208 * 72;        // 8 x [16][68]
  float* Pw = Qw + 8 * 16 * 68;     // 8 x [16][20]

  int bh = blockIdx.x; int b = bh >> 3, h = bh & 7;
  int tid = threadIdx.x;
  const float* kb = k + ((size_t)b * 196) * 512 + h * 64;
  const float* vb = v + ((size_t)b * 196) * 512 + h * 64;
  for (int idx = tid; idx < 208 * 16; idx += 256) {
    int r = idx >> 4, jc = (idx & 15) << 2;
    float4 kv = {0.f,0.f,0.f,0.f}, vv = {0.f,0.f,0.f,0.f};
    if (r < 196) {
      kv = *(const float4*)(kb + (size_t)r * 512 + jc);
      vv = *(const float4*)(vb + (size_t)r * 512 + jc);
    }
    *(float4*)(Kh + r * 68 + jc) = kv;
    *(float4*)(Vs + r * 72 + jc) = vv;
  }
  __syncthreads();

  int wid = tid >> 5, lane = tid & 31, half = lane >> 4, l15 = lane & 15;
  float* Qs = Qw + wid * 16 * 68;
  float* Pb = Pw + wid * 16 * 20;
  const float* qb = q + ((size_t)b * 196) * 512 + h * 64;

  for (int qt = wid; qt < 13; qt += 8) {
    // load this wave's Q tile (zero-padded)
    for (int i = lane; i < 256; i += 32) {
      int r = i >> 4, jc = (i & 15) << 2;
      int qrow = qt * 16 + r;
      float4 qv = {0.f,0.f,0.f,0.f};
      if (qrow < 196) qv = *(const float4*)(qb + (size_t)qrow * 512 + jc);
      *(float4*)(Qs + r * 68 + jc) = qv;
    }
    v8f accS[13];
#pragma unroll
    for (int kt = 0; kt < 13; kt++) {
      if (kt <= qt) {
        v8f s = {0,0,0,0,0,0,0,0};
#pragma unroll
        for (int d4 = 0; d4 < 64; d4 += 4) {
          int dk = d4 + 2 * half;
          v2f a;  a.x  = Qs[l15 * 68 + dk];                a.y  = Qs[l15 * 68 + dk + 1];
          v2f bb; bb.x = Kh[(kt * 16 + l15) * 68 + dk];    bb.y = Kh[(kt * 16 + l15) * 68 + dk + 1];
          s = wmma4(a, bb, s);
        }
        accS[kt] = s;
      }
    }
    // masked softmax (scale 64^-0.5 = 0.125); row stats across 16-lane groups
    float inv8[8];
#pragma unroll
    for (int vv_ = 0; vv_ < 8; vv_++) {
      int row = qt * 16 + vv_ + 8 * half;
      float mx = -3.0e38f;
#pragma unroll
      for (int kt = 0; kt < 13; kt++) {
        if (kt <= qt) {
          int col = kt * 16 + l15;
          float sv = accS[kt][vv_] * 0.125f;
          bool ok = (col <= row) && (col < 196) && (row < 196);
          sv = ok ? sv : -3.0e38f;
          accS[kt][vv_] = sv;
          mx = fmaxf(mx, sv);
        }
      }
      for (int m = 1; m < 16; m <<= 1) mx = fmaxf(mx, __shfl_xor(mx, m, 32));
      float sum = 0.f;
#pragma unroll
      for (int kt = 0; kt < 13; kt++) {
        if (kt <= qt) {
          float sv = accS[kt][vv_];
          float p = (sv > -1.0e38f) ? expf(sv - mx) : 0.0f;
          accS[kt][vv_] = p;
          sum += p;
        }
      }
      for (int m = 1; m < 16; m <<= 1) sum += __shfl_xor(sum, m, 32);
      inv8[vv_] = (sum > 0.f) ? 1.0f / sum : 0.0f;
    }
    // O = P @ V : restage each P tile as A-fragments through per-wave LDS
    v8f accO[4];
#pragma unroll
    for (int nt = 0; nt < 4; nt++) { v8f z = {0,0,0,0,0,0,0,0}; accO[nt] = z; }
#pragma unroll
    for (int kt = 0; kt < 13; kt++) {
      if (kt <= qt) {
#pragma unroll
        for (int vv_ = 0; vv_ < 8; vv_++)
          Pb[(vv_ + 8 * half) * 20 + l15] = accS[kt][vv_];
        // same-wave LDS ops are in-order; compiler inserts dscnt waits
#pragma unroll
        for (int ks = 0; ks < 16; ks += 4) {
          int kk = ks + 2 * half;
          v2f a; a.x = Pb[l15 * 20 + kk]; a.y = Pb[l15 * 20 + kk + 1];
#pragma unroll
          for (int nt = 0; nt < 4; nt++) {
            v2f bb;
            bb.x = Vs[(kt * 16 + kk) * 72 + nt * 16 + l15];
            bb.y = Vs[(kt * 16 + kk + 1) * 72 + nt * 16 + l15];
            accO[nt] = wmma4(a, bb, accO[nt]);
          }
        }
      }
    }
#pragma unroll
    for (int nt = 0; nt < 4; nt++) {
#pragma unroll
      for (int vv_ = 0; vv_ < 8; vv_++) {
        int row = qt * 16 + vv_ + 8 * half;
        if (row < 196)
          o[((size_t)b * 196 + row) * 512 + h * 64 + nt * 16 + l15] = accO[nt][vv_] * inv8[vv_];
      }
    }
  }
}

// ---------------------------------------------------------------------------
// Router noisy logits: noisy = x@rtW + rtb + noise * softplus(x@rnW + rnb)
// 32 tokens per block; thread = (token_local, expert)
// ---------------------------------------------------------------------------
__global__ __launch_bounds__(256) void router_logits_kernel(
    const float* __restrict__ x, const float* __restrict__ rtW, const float* __restrict__ rtb,
    const float* __restrict__ rnW, const float* __restrict__ rnb,
    const float* __restrict__ noise, float* __restrict__ noisy)
{
  int tid = threadIdx.x;
  int token = blockIdx.x * 32 + (tid >> 3);
  int e = tid & 7;
  const float* xr = x + (size_t)token * 512;
  float at = 0.f, an = 0.f;
  for (int d = 0; d < 512; d++) {
    float xv = xr[d];
    at += xv * rtW[d * 8 + e];
    an += xv * rnW[d * 8 + e];
  }
  at += rtb[e]; an += rnb[e];
  float sp = fmaxf(an, 0.f) + log1pf(expf(-fabsf(an)));   // stable softplus
  noisy[(size_t)token * 8 + e] = at + noise[(size_t)token * 8 + e] * sp;
}

// Top-2 selection (jax tie rule: earliest index) + softmax over selected
__global__ __launch_bounds__(256) void gate_kernel(
    const float* __restrict__ noisy, float* __restrict__ gate, int M)
{
  int token = blockIdx.x * 256 + threadIdx.x;
  if (token >= M) return;
  float n[8];
#pragma unroll
  for (int e = 0; e < 8; e++) n[e] = noisy[(size_t)token * 8 + e];
  int i1 = 0; float v1 = n[0];
#pragma unroll
  for (int e = 1; e < 8; e++) if (n[e] > v1) { v1 = n[e]; i1 = e; }
  int i2 = -1; float v2 = -3.0e38f;
#pragma unroll
  for (int e = 0; e < 8; e++) if (e != i1 && n[e] > v2) { v2 = n[e]; i2 = e; }
  float e2 = expf(v2 - v1);
  float s = 1.0f + e2;
#pragma unroll
  for (int e = 0; e < 8; e++) gate[(size_t)token * 8 + e] = 0.f;
  gate[(size_t)token * 8 + i1] = 1.0f / s;
  gate[(size_t)token * 8 + i2] = e2 / s;
}

// ---------------------------------------------------------------------------
// Fused MoE: y = sum_e (gate_e ⊙rows relu(x Wi_e + bi_e)) Wo_e + sum_e gate_e bo_e
// 32-token tile per WG, 8 waves. FF chunked by 64 through double-buffered LDS H.
// H phase: wave (mw=wid&1, ht=wid>>1) -> 16x16 subtile, K=512 (128 WMMA)
// Y phase: wave (mw, nb=wid>>1) -> 8 n-tiles x (K=64) (128 WMMA)
// ---------------------------------------------------------------------------
__global__ __launch_bounds__(256) void moe_kernel(
    const float* __restrict__ x, const float* __restrict__ gate,
    const float* __restrict__ Wi, const float* __restrict__ bi,
    const float* __restrict__ Wo, const float* __restrict__ bo,
    float* __restrict__ y)
{
  extern __shared__ float smem[];
  float* xs  = smem;                  // [32][516]
  float* Hb  = xs + 32 * 516;         // [2][32][68]
  float* gts = Hb + 2 * 32 * 68;      // [32][8]
  int tid = threadIdx.x;
  int m0 = blockIdx.x * 32;
  for (int idx = tid; idx < 32 * 128; idx += 256) {
    int r = idx >> 7, jc = (idx & 127) << 2;
    float4 t = *(const float4*)(x + ((size_t)(m0 + r)) * 512 + jc);
    *(float4*)(xs + r * 516 + jc) = t;
  }
  gts[tid] = gate[(size_t)m0 * 8 + tid];   // 32*8 == 256
  __syncthreads();

  int wid = tid >> 5, lane = tid & 31, half = lane >> 4, l15 = lane & 15;
  int mw = wid & 1, ht = wid >> 1, nb = wid >> 1;
  v8f accY[8];
#pragma unroll
  for (int nt = 0; nt < 8; nt++) { v8f z = {0,0,0,0,0,0,0,0}; accY[nt] = z; }

  for (int e = 0; e < 8; e++) {
    float anyg = 0.f;
#pragma unroll
    for (int r = 0; r < 32; r++) anyg += gts[r * 8 + e];   // gates >= 0, uniform
    if (anyg <= 0.f) continue;
    const float* WiE = Wi + (size_t)e * 512 * 2048;
    const float* WoE = Wo + (size_t)e * 2048 * 512;
    const float* biE = bi + e * 2048;
    for (int c = 0; c < 32; c++) {
      float* HbB = Hb + (c & 1) * 32 * 68;
      // ---- H phase ----
      v8f hacc = {0,0,0,0,0,0,0,0};
#pragma unroll 8
      for (int k4 = 0; k4 < 512; k4 += 4) {
        int kk = k4 + 2 * half;
        v2f a;  a.x  = xs[(mw * 16 + l15) * 516 + kk];  a.y = xs[(mw * 16 + l15) * 516 + kk + 1];
        v2f bb;
        bb.x = WiE[(size_t)kk * 2048 + c * 64 + ht * 16 + l15];
        bb.y = WiE[(size_t)(kk + 1) * 2048 + c * 64 + ht * 16 + l15];
        hacc = wmma4(a, bb, hacc);
      }
#pragma unroll
      for (int vv_ = 0; vv_ < 8; vv_++) {
        int r = mw * 16 + vv_ + 8 * half;
        int colh = c * 64 + ht * 16 + l15;
        float hv = hacc[vv_] + biE[colh];
        hv = fmaxf(hv, 0.f) * gts[r * 8 + e];
        HbB[r * 68 + ht * 16 + l15] = hv;
      }
      __syncthreads();       // single barrier; double buffer covers WAR
      // ---- Y phase ----
#pragma unroll
      for (int nt = 0; nt < 8; nt++) {
        int ncol = nb * 128 + nt * 16 + l15;
#pragma unroll
        for (int k4 = 0; k4 < 64; k4 += 4) {
          int kk = k4 + 2 * half;
          v2f a; a.x = HbB[(mw * 16 + l15) * 68 + kk]; a.y = HbB[(mw * 16 + l15) * 68 + kk + 1];
          v2f bb;
          bb.x = WoE[(size_t)(c * 64 + kk) * 512 + ncol];
          bb.y = WoE[(size_t)(c * 64 + kk + 1) * 512 + ncol];
          accY[nt] = wmma4(a, bb, accY[nt]);
        }
      }
    }
  }
#pragma unroll
  for (int nt = 0; nt < 8; nt++) {
#pragma unroll
    for (int vv_ = 0; vv_ < 8; vv_++) {
      int r = mw * 16 + vv_ + 8 * half;
      int col = nb * 128 + nt * 16 + l15;
      float val = accY[nt][vv_];
#pragma unroll
      for (int e = 0; e < 8; e++) val += gts[r * 8 + e] * bo[e * 512 + col];
      y[((size_t)(m0 + r)) * 512 + col] = val;
    }
  }
}

// LayerNorm over last dim (512), two-pass like the reference
__global__ __launch_bounds__(256) void ln_kernel(
    const float* __restrict__ x, const float* __restrict__ g,
    const float* __restrict__ bta, float* __restrict__ out)
{
  __shared__ float red[8];
  int row = blockIdx.x, tid = threadIdx.x;
  const float* xr = x + (size_t)row * 512;
  float a = xr[tid], b = xr[tid + 256];
  float s = a + b;
  for (int m = 1; m < 32; m <<= 1) s += __shfl_xor(s, m, 32);
  if ((tid & 31) == 0) red[tid >> 5] = s;
  __syncthreads();
  float mu = 0.f;
#pragma unroll
  for (int i = 0; i < 8; i++) mu += red[i];
  mu *= (1.0f / 512.0f);
  __syncthreads();
  float d1 = a - mu, d2 = b - mu;
  float qv = d1 * d1 + d2 * d2;
  for (int m = 1; m < 32; m <<= 1) qv += __shfl_xor(qv, m, 32);
  if ((tid & 31) == 0) red[tid >> 5] = qv;
  __syncthreads();
  float var = 0.f;
#pragma unroll
  for (int i = 0; i < 8; i++) var += red[i];
  var *= (1.0f / 512.0f);
  float rs = rsqrtf(var + 1e-5f);
  out[(size_t)row * 512 + tid]       = d1 * rs * g[tid] + bta[tid];
  out[(size_t)row * 512 + tid + 256] = d2 * rs * g[tid + 256] + bta[tid + 256];
}

// global_vector[b,d] = sum_t second_vector[b,t,d]
__global__ __launch_bounds__(256) void gsum_kernel(
    const float* __restrict__ sv, float* __restrict__ out)
{
  int b = blockIdx.x;
  int d = blockIdx.y * 256 + threadIdx.x;
  float s = 0.f;
  for (int t = 0; t < 196; t++) s += sv[((size_t)b * 196 + t) * 512 + d];
  out[(size_t)b * 512 + d] = s;
}

// ---------------------------------------------------------------------------
extern "C" void kernel_launch(void* const* d_in, const int* in_sizes, int n_in,
                              void* d_out, int out_size, void* d_ws, size_t ws_size,
                              hipStream_t stream) {
  (void)in_sizes; (void)n_in; (void)out_size; (void)ws_size;
  const float* P[43];
  for (int i = 0; i < 43; i++) P[i] = (const float*)d_in[i];
  const float* ximg = P[0];
  const float* pembW = P[1]; const float* pembB = P[2];
  const float* vecW  = P[3]; const float* vecB  = P[4];
  const float* clsW  = P[5]; const float* clsB  = P[6];

  float* out = (float*)d_out;
  float* ws  = (float*)d_ws;
  const int M = NTOK;
  const size_t BIG = (size_t)M * DMODEL;
  float* bA = ws;
  float* bB = bA + BIG;
  float* bC = bB + BIG;
  float* bD = bC + BIG;
  float* bE = bD + BIG;
  float* nz = bE + BIG;
  float* gt = nz + (size_t)M * 8;

  dim3 blk(256);
  dim3 gFull(8, 98);   // N=512 -> 8 col tiles, M=6272 -> 98 row tiles
  const size_t ATTN_SMEM = (size_t)(208*68 + 208*72 + 8*16*68 + 8*16*20) * 4;
  const size_t MOE_SMEM  = (size_t)(32*516 + 2*32*68 + 32*8) * 4;

  // stage 0: patchify + patch embedding + posenc  -> bA
  gemm_wmma_kernel<<<gFull, blk, 0, stream>>>(ximg, pembW, pembB, bA, M, 512, 256, 2, 1);

  const float* hin = bA;
  float* fv[2] = { out, out + BIG };
  for (int l = 0; l < 2; l++) {
    int base = 7 + l * 18;
    const float* pW  = P[base + 0]; const float* pb  = P[base + 1];
    const float* Wq  = P[base + 2]; const float* Wk  = P[base + 3];
    const float* Wv  = P[base + 4]; const float* Wo  = P[base + 5];
    const float* bo  = P[base + 6];
    const float* rtW = P[base + 7]; const float* rtb = P[base + 8];
    const float* rnW = P[base + 9]; const float* rnb = P[base + 10];
    const float* eWi = P[base + 11]; const float* ebi = P[base + 12];
    const float* eWo = P[base + 13]; const float* ebo = P[base + 14];
    const float* lng = P[base + 15]; const float* lnb = P[base + 16];
    const float* noi = P[base + 17];

    gemm_wmma_kernel<<<gFull, blk, 0, stream>>>(hin, pW, pb, bB, M, 512, 512, 1, 0);       // xp
    gemm_wmma_kernel<<<gFull, blk, 0, stream>>>(bB, Wq, nullptr, bC, M, 512, 512, 0, 0);   // q
    gemm_wmma_kernel<<<gFull, blk, 0, stream>>>(bB, Wk, nullptr, bD, M, 512, 512, 0, 0);   // k
    gemm_wmma_kernel<<<gFull, blk, 0, stream>>>(bB, Wv, nullptr, bE, M, 512, 512, 0, 0);   // v
    attn_kernel<<<dim3(256), blk, ATTN_SMEM, stream>>>(bC, bD, bE, bB);                    // attnout
    gemm_wmma_kernel<<<gFull, blk, 0, stream>>>(bB, Wo, bo, bC, M, 512, 512, 1, 0);        // att
    router_logits_kernel<<<dim3(196), blk, 0, stream>>>(bC, rtW, rtb, rnW, rnb, noi, nz);
    gate_kernel<<<dim3(25), blk, 0, stream>>>(nz, gt, M);
    moe_kernel<<<dim3(196), blk, MOE_SMEM, stream>>>(bC, gt, eWi, ebi, eWo, ebo, bD);      // ymoe
    ln_kernel<<<dim3(M), blk, 0, stream>>>(bD, lng, lnb, bE);                              // f
    gemm_wmma_kernel<<<gFull, blk, 0, stream>>>(bE, vecW, vecB, fv[l], M, 512, 512, 1, 0); // *_vector
    hin = fv[l];
  }

  float* gv  = out + 2 * BIG;           // global_vector [32,512]
  float* cls = gv + 32 * 512;           // cls_vector    [32,512]
  gsum_kernel<<<dim3(32, 2), blk, 0, stream>>>(out + BIG, gv);
  gemm_wmma_kernel<<<dim3(8, 1), blk, 0, stream>>>(gv, clsW, clsB, cls, 32, 512, 512, 1, 0);
}